// DeformConv2dBlock_84670985274136
// MI455X (gfx1250) — compile-verified
//
#include <hip/hip_runtime.h>

typedef __attribute__((ext_vector_type(2))) float v2f;
typedef __attribute__((ext_vector_type(8))) float v8f;

namespace {
constexpr int B_ = 4, C_ = 64, H_ = 256, W_ = 256;
constexpr int K_ = 9;                       // 3x3 taps
constexpr int HO_ = H_ - 2, WO_ = W_ - 2;   // 254
constexpr int OCH_ = 2 * K_;                // 18 offset channels
constexpr int KKA_ = C_ * K_;               // 576 reduction dim (offset conv)
constexpr int KKP_ = 580;                   // padded LDS stride (bank-spread)
constexpr int NPIX_O = HO_ * WO_;           // 64516
constexpr int NPIX_F = H_ * W_;             // 65536
}

__device__ __forceinline__ v8f wmma_f32_16x16x4(v2f a, v2f b, v8f c) {
  // D(16x16) = A(16x4) * B(4x16) + C, fp32 WMMA (wave32)
  return __builtin_amdgcn_wmma_f32_16x16x4_f32(false, a, false, b, (short)0, c,
                                               false, false);
}

// ---------------------------------------------------------------------------
// Kernel A: offset = conv3x3(x, offset_w) + offset_b   via WMMA GEMM
//   D tile per wave: M = 32 (18 valid offset chans), N = 32 pixels (2 tiles)
//   A (32 x 576) staged in LDS zero-padded; B = im2col (576 x pixels).
//   Per K-step: 4 ds_load + 4 global_load -> 4 v_wmma_f32_16x16x4_f32.
// ---------------------------------------------------------------------------
__global__ void offset_conv_wmma(const float* __restrict__ x,
                                 const float* __restrict__ ow,
                                 const float* __restrict__ ob,
                                 float* __restrict__ off) {
  __shared__ float wlds[32 * KKP_];  // ~74 KB of the 320 KB WGP LDS
  // Cooperative stage: 18 real rows + 14 zero rows (kills branches in loop).
  for (int i = threadIdx.x; i < 32 * KKA_; i += 256) {
    const int row = i / KKA_;
    const int col = i - row * KKA_;
    wlds[row * KKP_ + col] = (row < OCH_) ? ow[row * KKA_ + col] : 0.0f;
  }
  __syncthreads();

  const int lane = threadIdx.x & 31;
  const int wave = threadIdx.x >> 5;
  const int tile = blockIdx.x * 8 + wave;          // 32-pixel macro tile
  const int ntiles = (NPIX_O + 31) / 32;           // 2017
  if (tile >= ntiles) return;                      // wave-uniform
  const int b = blockIdx.y;
  const int half = lane >> 4;
  const int lr = lane & 15;

  const int p0 = tile * 32 + lr;
  const int p1 = p0 + 16;
  const bool v0 = p0 < NPIX_O, v1 = p1 < NPIX_O;
  const int pc0 = v0 ? p0 : (NPIX_O - 1);
  const int pc1 = v1 ? p1 : (NPIX_O - 1);
  const int ho0 = pc0 / WO_, wo0 = pc0 - ho0 * WO_;
  const int ho1 = pc1 / WO_, wo1 = pc1 - ho1 * WO_;
  const float* xb = x + (size_t)b * C_ * H_ * W_;
  const float* wrow0 = wlds + lr * KKP_;
  const float* wrow1 = wlds + (lr + 16) * KKP_;

  v8f acc00 = {}, acc10 = {}, acc01 = {}, acc11 = {};
  for (int step = 0; step < KKA_ / 4; ++step) {
    const int kk0 = step * 4 + half * 2;
    const int kk1 = kk0 + 1;
    v2f a0, a1, b0, b1;
    a0.x = wrow0[kk0];  a0.y = wrow0[kk1];   // conflict-free ds_load
    a1.x = wrow1[kk0];  a1.y = wrow1[kk1];
    {
      const int c = kk0 / 9, r = kk0 - c * 9;
      const int base = (c * H_ + r / 3) * W_ + (r % 3);
      b0.x = xb[base + ho0 * W_ + wo0];
      b1.x = xb[base + ho1 * W_ + wo1];
    }
    {
      const int c = kk1 / 9, r = kk1 - c * 9;
      const int base = (c * H_ + r / 3) * W_ + (r % 3);
      b0.y = xb[base + ho0 * W_ + wo0];
      b1.y = xb[base + ho1 * W_ + wo1];
    }
    acc00 = wmma_f32_16x16x4(a0, b0, acc00);
    acc10 = wmma_f32_16x16x4(a1, b0, acc10);
    acc01 = wmma_f32_16x16x4(a0, b1, acc01);
    acc11 = wmma_f32_16x16x4(a1, b1, acc11);
  }

  float* offb = off + (size_t)b * OCH_ * NPIX_O;
  if (v0) {
#pragma unroll
    for (int i = 0; i < 8; ++i) {
      const int och = i + 8 * half;
      offb[och * NPIX_O + p0] = acc00[i] + ob[och];
      if (och + 16 < OCH_) offb[(och + 16) * NPIX_O + p0] = acc10[i] + ob[och + 16];
    }
  }
  if (v1) {
#pragma unroll
    for (int i = 0; i < 8; ++i) {
      const int och = i + 8 * half;
      offb[och * NPIX_O + p1] = acc01[i] + ob[och];
      if (och + 16 < OCH_) offb[(och + 16) * NPIX_O + p1] = acc11[i] + ob[och + 16];
    }
  }
}

// ---------------------------------------------------------------------------
// Kernel B: P[b,k,y,x] = sum_c deform_w[0,c,k] * x[b,c,y,x]  (1x1 conv, WMMA)
//   D tile per wave: M = 9 taps (zero-padded to 16), N = 32 pixels (2 tiles)
// ---------------------------------------------------------------------------
__global__ void proj_wmma(const float* __restrict__ x,
                          const float* __restrict__ dw,
                          float* __restrict__ P) {
  __shared__ float wlds[C_ * 16];  // 4 KB, transposed+zero-padded deform_w
  for (int i = threadIdx.x; i < C_ * 16; i += 256) {
    const int c = i >> 4, r = i & 15;
    wlds[i] = (r < K_) ? dw[c * K_ + r] : 0.0f;
  }
  __syncthreads();

  const int lane = threadIdx.x & 31;
  const int wave = threadIdx.x >> 5;
  const int tile = blockIdx.x * 8 + wave;  // 32-pixel macro tile
  const int b = blockIdx.y;
  const int half = lane >> 4;
  const int lr = lane & 15;
  const int p0 = tile * 32 + lr;           // NPIX_F % 32 == 0: always valid
  const int p1 = p0 + 16;
  const int y0 = p0 >> 8, x0 = p0 & (W_ - 1);
  const int y1 = p1 >> 8, x1 = p1 & (W_ - 1);
  const float* xb = x + (size_t)b * C_ * H_ * W_;

  v8f acc0 = {}, acc1 = {};
  for (int step = 0; step < C_ / 4; ++step) {
    const int c0 = step * 4 + half * 2;
    v2f af, b0, b1;
    af.x = wlds[c0 * 16 + lr];
    af.y = wlds[(c0 + 1) * 16 + lr];
    b0.x = xb[(c0 * H_ + y0) * W_ + x0];         // coalesced over pixels
    b0.y = xb[((c0 + 1) * H_ + y0) * W_ + x0];
    b1.x = xb[(c0 * H_ + y1) * W_ + x1];
    b1.y = xb[((c0 + 1) * H_ + y1) * W_ + x1];
    acc0 = wmma_f32_16x16x4(af, b0, acc0);
    acc1 = wmma_f32_16x16x4(af, b1, acc1);
  }
  float* Pb = P + (size_t)b * K_ * NPIX_F;
#pragma unroll
  for (int i = 0; i < 8; ++i) {
    const int k = i + 8 * half;
    if (k < K_) {
      Pb[k * NPIX_F + p0] = acc0[i];
      Pb[k * NPIX_F + p1] = acc1[i];
    }
  }
}

// ---------------------------------------------------------------------------
// Kernel C: bilinear gather on the 9-channel projected image + bias
// ---------------------------------------------------------------------------
__global__ void deform_gather(const float* __restrict__ off,
                              const float* __restrict__ P,
                              const float* __restrict__ db,
                              float* __restrict__ out, int total) {
  const int idx = blockIdx.x * blockDim.x + threadIdx.x;
  if (idx >= total) return;
  const int b = idx / NPIX_O;
  const int p = idx - b * NPIX_O;
  const int ho = p / WO_;
  const int wo = p - ho * WO_;
  const float* offb = off + (size_t)b * OCH_ * NPIX_O;
  const float* Pb = P + (size_t)b * K_ * NPIX_F;
  float acc = db[0];
#pragma unroll
  for (int k = 0; k < K_; ++k) {
    const float dy = offb[(2 * k) * NPIX_O + p];
    const float dx = offb[(2 * k + 1) * NPIX_O + p];
    const float py = (float)(ho + k / 3) + dy;
    const float px = (float)(wo + (k % 3)) + dx;
    const float y0f = floorf(py), x0f = floorf(px);
    const float ly = py - y0f, lx = px - x0f;
    const int y0 = (int)y0f, x0 = (int)x0f;
    const float* Pk = Pb + k * NPIX_F;
    const float w00 = (1.f - ly) * (1.f - lx);
    const float w01 = (1.f - ly) * lx;
    const float w10 = ly * (1.f - lx);
    const float w11 = ly * lx;
    if ((unsigned)y0 < (unsigned)H_ && (unsigned)x0 < (unsigned)W_)
      acc += w00 * Pk[y0 * W_ + x0];
    if ((unsigned)y0 < (unsigned)H_ && (unsigned)(x0 + 1) < (unsigned)W_)
      acc += w01 * Pk[y0 * W_ + x0 + 1];
    if ((unsigned)(y0 + 1) < (unsigned)H_ && (unsigned)x0 < (unsigned)W_)
      acc += w10 * Pk[(y0 + 1) * W_ + x0];
    if ((unsigned)(y0 + 1) < (unsigned)H_ && (unsigned)(x0 + 1) < (unsigned)W_)
      acc += w11 * Pk[(y0 + 1) * W_ + x0 + 1];
  }
  out[idx] = acc;
}

// ---------------------------------------------------------------------------
extern "C" void kernel_launch(void* const* d_in, const int* in_sizes, int n_in,
                              void* d_out, int out_size, void* d_ws,
                              size_t ws_size, hipStream_t stream) {
  (void)in_sizes; (void)n_in; (void)out_size; (void)ws_size;
  const float* x  = (const float*)d_in[0];  // (4,64,256,256)
  const float* ow = (const float*)d_in[1];  // (18,64,3,3)
  const float* ob = (const float*)d_in[2];  // (18,)
  const float* dw = (const float*)d_in[3];  // (1,64,3,3)
  const float* db = (const float*)d_in[4];  // (1,)
  float* out = (float*)d_out;               // (4,1,254,254)

  char* ws = (char*)d_ws;
  float* off_buf = (float*)ws;                                          // 18.6 MB
  float* P = (float*)(ws + (size_t)B_ * OCH_ * NPIX_O * sizeof(float)); // 9.4 MB

  // Kernel A: offset conv (WMMA f32). 8 waves/block, 32 pixels per wave.
  const int tilesA = (NPIX_O + 31) / 32;  // 2017
  dim3 grdA((tilesA + 7) / 8, B_);
  offset_conv_wmma<<<grdA, dim3(256), 0, stream>>>(x, ow, ob, off_buf);

  // Kernel B: channel projection (WMMA f32). 32 pixels per wave.
  const int tilesB = NPIX_F / 32;  // 2048
  dim3 grdB(tilesB / 8, B_);
  proj_wmma<<<grdB, dim3(256), 0, stream>>>(x, dw, P);

  // Kernel C: bilinear gather on 9-channel projection.
  const int total = B_ * NPIX_O;
  deform_gather<<<(total + 255) / 256, 256, 0, stream>>>(off_buf, P, db, out,
                                                         total);
}